// RecGRU_W_42691974922285
// MI455X (gfx1250) — compile-verified
//
#include <hip/hip_runtime.h>
#include <hip/hip_bf16.h>

typedef __attribute__((ext_vector_type(2))) float v2f;
typedef __attribute__((ext_vector_type(8))) float v8f;

#define DFEAT 64
#define APAD 68   // LDS row stride (floats): 4*row + k0 -> 64 distinct banks per wave

// ---------------------------------------------------------------- utilities

__global__ void zero_u32_kernel(unsigned* p, int n) {
    int i = blockIdx.x * blockDim.x + threadIdx.x;
    if (i < n) p[i] = 0u;
}

// deg[src] += w  (segment_sum over src)
__global__ void deg_kernel(const int* __restrict__ ei, const float* __restrict__ ew,
                           float* __restrict__ deg, int E) {
    int e = blockIdx.x * blockDim.x + threadIdx.x;
    if (e < E) atomicAdd(&deg[ei[e]], ew[e]);
}

__global__ void dinv_kernel(float* __restrict__ deg, int N) {
    int i = blockIdx.x * blockDim.x + threadIdx.x;
    if (i < N) {
        float d = deg[i];
        deg[i] = (d > 0.0f) ? rsqrtf(d) : 0.0f;
    }
}

// histogram of dst
__global__ void count_kernel(const int* __restrict__ ei, unsigned* __restrict__ cnt,
                             int E) {
    int e = blockIdx.x * blockDim.x + threadIdx.x;
    if (e < E) atomicAdd(&cnt[ei[E + e]], 1u);
}

// single-block exclusive scan over cnt[0..N-1] -> rowptr[0..N], rownext copy
__global__ void scan_kernel(const unsigned* __restrict__ cnt,
                            unsigned* __restrict__ rowptr,
                            unsigned* __restrict__ rownext, int N, int E) {
    __shared__ unsigned buf[1024];
    __shared__ unsigned carry;
    int tid = threadIdx.x;
    if (tid == 0) carry = 0u;
    __syncthreads();
    for (int base = 0; base < N; base += 1024) {
        int i = base + tid;
        unsigned x = (i < N) ? cnt[i] : 0u;
        buf[tid] = x;
        __syncthreads();
        for (int off = 1; off < 1024; off <<= 1) {
            unsigned t = (tid >= off) ? buf[tid - off] : 0u;
            __syncthreads();
            buf[tid] += t;
            __syncthreads();
        }
        unsigned incl = buf[tid];
        unsigned c = carry;
        if (i < N) { unsigned ex = c + incl - x; rowptr[i] = ex; rownext[i] = ex; }
        __syncthreads();
        if (tid == 1023) carry = c + incl;
        __syncthreads();
    }
    if (tid == 0) rowptr[N] = (unsigned)E;
}

// scatter edges into CSR-by-dst; fuse symmetric normalization; pack {col,val}
__global__ void fill_kernel(const int* __restrict__ ei, const float* __restrict__ ew,
                            const float* __restrict__ dinv,
                            unsigned* __restrict__ rownext,
                            uint2* __restrict__ csr, int E) {
    int e = blockIdx.x * blockDim.x + threadIdx.x;
    if (e < E) {
        int s = ei[e];
        int d = ei[E + e];
        float nv = ew[e] * dinv[s] * dinv[d];
        unsigned p = atomicAdd(&rownext[d], 1u);
        csr[p] = make_uint2((unsigned)s, __float_as_uint(nv));
    }
}

// Fused gather-SpMM + Chebyshev recurrence, 2 features (float2) per thread.
// order2==0: out = (c-1)*vin - c*(A vin)                      (= l_hat(vin))
// order2==1: out = 2*((c-1)*vin - c*(A vin)) - prev           (= 2 l_hat - T_{k-1})
__global__ void cheb_spmm_kernel(const float* __restrict__ vin,
                                 const float* __restrict__ prev,
                                 float* __restrict__ out,
                                 const unsigned* __restrict__ rowptr,
                                 const uint2* __restrict__ csr,
                                 const float* __restrict__ lam,
                                 int N, int order2) {
    int t = blockIdx.x * blockDim.x + threadIdx.x;
    int n = t >> 5;           // node
    int fp = t & 31;          // float2 feature pair (64 feats = 32 pairs)
    if (n >= N) return;
    float c = 2.0f / lam[0];
    const float2* vin2 = (const float2*)vin;
    unsigned beg = rowptr[n], end = rowptr[n + 1];
    float ax = 0.0f, ay = 0.0f;
    for (unsigned j = beg; j < end; ++j) {
        uint2 cv = csr[j];                       // one b64 broadcast per edge
        float w = __uint_as_float(cv.y);
        float2 v = vin2[(size_t)cv.x * 32 + fp]; // coalesced 256B/row per wave
        ax = fmaf(w, v.x, ax);
        ay = fmaf(w, v.y, ay);
    }
    size_t g = (size_t)n * 32 + fp;
    float2 vg = vin2[g];
    float lx = (c - 1.0f) * vg.x - c * ax;
    float ly = (c - 1.0f) * vg.y - c * ay;
    if (order2) {
        float2 pv = ((const float2*)prev)[g];
        lx = 2.0f * lx - pv.x;
        ly = 2.0f * ly - pv.y;
    }
    ((float2*)out)[g] = make_float2(lx, ly);
}

// ------------------------------------------------- fused WMMA gate kernels
// A tiles staged in LDS (row stride APAD -> conflict-free b64 fragment reads).
// One wave computes one 16x16 fp32 tile via V_WMMA_F32_16X16X4_F32.

template <int NT>
__device__ __forceinline__ void stage_tiles(float (*As)[16][APAD],
                                            const float* const* Ab,
                                            int r0, int N, int tid) {
    #pragma unroll
    for (int tile = 0; tile < 6; ++tile) {
        const float2* src = (const float2*)Ab[tile];
        #pragma unroll
        for (int it = 0; it < 512 / NT; ++it) {       // 16*32 float2 per tile
            int idx = tid + it * NT;
            int row = idx >> 5;
            int cp = idx & 31;
            int rr = r0 + row;
            rr = (rr < N) ? rr : (N - 1);             // clamp: no branches
            float2 v = src[(size_t)rr * 32 + cp];
            *(float2*)&As[tile][row][cp * 2] = v;
        }
    }
    __syncthreads();
}

__device__ __forceinline__ v8f wmma_accum_terms(const float (*As)[16][APAD],
                                                const float* __restrict__ cw,
                                                int ci_base, int c0, int lane) {
    int mrow = lane & 15;
    int khalf = (lane >> 4) << 1;     // 0 or 2
    int ncol = lane & 15;
    v8f acc = {};
    #pragma unroll
    for (int s = 0; s < 2; ++s) {
        int ci = ci_base + s;
        #pragma unroll
        for (int k = 0; k < 3; ++k) {
            const float* arow = &As[s * 3 + k][mrow][0];
            const float* Wc = cw + (size_t)(ci * 3 + k) * (DFEAT * DFEAT) + c0 + ncol;
            #pragma unroll
            for (int kb = 0; kb < 16; ++kb) {
                int k0 = kb * 4 + khalf;
                v2f a = *(const v2f*)(arow + k0);     // LDS b64, bank-conflict-free
                v2f b;
                b.x = Wc[(size_t)k0 * DFEAT];
                b.y = Wc[(size_t)k0 * DFEAT + DFEAT];
                acc = __builtin_amdgcn_wmma_f32_16x16x4_f32(
                    false, a, false, b, (short)0, acc, false, false);
            }
        }
    }
    return acc;
}

// Stage C: Z = sigmoid(cheb(X;W0)+cheb(H;W1)), R = sigmoid(cheb(X;W2)+cheb(H;W3)),
//          HR = H * R.  8 waves: wave&3 = col tile, wave>>2 = gate.
__global__ void gates_kernel(const float* __restrict__ X,
                             const float* __restrict__ TX1, const float* __restrict__ TX2,
                             const float* __restrict__ H,
                             const float* __restrict__ TH1, const float* __restrict__ TH2,
                             const float* __restrict__ cw, const float* __restrict__ cb,
                             float* __restrict__ Zb, float* __restrict__ HRb, int N) {
    __shared__ float As[6][16][APAD];
    int lane = threadIdx.x & 31;
    int wave = threadIdx.x >> 5;
    int c0 = (wave & 3) * 16;
    int gate = wave >> 2;           // 0: Z (convs 0,1), 1: R (convs 2,3)
    int r0 = blockIdx.x * 16;

    const float* Ab[6] = {X, TX1, TX2, H, TH1, TH2};
    stage_tiles<256>(As, Ab, r0, N, threadIdx.x);

    v8f acc = wmma_accum_terms(As, cw, gate * 2, c0, lane);

    int ncol = lane & 15;
    float bias = cb[(gate * 2) * DFEAT + c0 + ncol] +
                 cb[(gate * 2 + 1) * DFEAT + c0 + ncol];
    int rbase = (lane >> 4) * 8;
    #pragma unroll
    for (int i = 0; i < 8; ++i) {
        int r = r0 + rbase + i;
        if (r < N) {
            size_t g = (size_t)r * DFEAT + c0 + ncol;
            float v = acc[i] + bias;
            float sg = 1.0f / (1.0f + __expf(-v));
            if (gate == 0) Zb[g] = sg;
            else           HRb[g] = H[g] * sg;
        }
    }
}

// Stage E: Ht = tanh(cheb(X;W4)+cheb(HR;W5)); out = Z*Ht + (1-Z)*H
// 4 waves, one col tile each.
__global__ void final_kernel(const float* __restrict__ X,
                             const float* __restrict__ TX1, const float* __restrict__ TX2,
                             const float* __restrict__ HR,
                             const float* __restrict__ THR1, const float* __restrict__ THR2,
                             const float* __restrict__ cw, const float* __restrict__ cb,
                             const float* __restrict__ Zb, const float* __restrict__ H,
                             float* __restrict__ out, int N) {
    __shared__ float As[6][16][APAD];
    int lane = threadIdx.x & 31;
    int wave = threadIdx.x >> 5;
    int c0 = wave * 16;
    int r0 = blockIdx.x * 16;

    const float* Ab[6] = {X, TX1, TX2, HR, THR1, THR2};
    stage_tiles<128>(As, Ab, r0, N, threadIdx.x);

    v8f acc = wmma_accum_terms(As, cw, 4, c0, lane);  // convs 4,5

    int ncol = lane & 15;
    float bias = cb[4 * DFEAT + c0 + ncol] + cb[5 * DFEAT + c0 + ncol];
    int rbase = (lane >> 4) * 8;
    #pragma unroll
    for (int i = 0; i < 8; ++i) {
        int r = r0 + rbase + i;
        if (r < N) {
            size_t g = (size_t)r * DFEAT + c0 + ncol;
            float ht = tanhf(acc[i] + bias);
            float z = Zb[g];
            out[g] = z * ht + (1.0f - z) * H[g];
        }
    }
}

// ---------------------------------------------------------------- launcher

extern "C" void kernel_launch(void* const* d_in, const int* in_sizes, int n_in,
                              void* d_out, int out_size, void* d_ws, size_t ws_size,
                              hipStream_t stream) {
    const float* X   = (const float*)d_in[0];
    const int*   ei  = (const int*)d_in[1];   // [2,E] int32 (JAX x64 disabled)
    const float* ew  = (const float*)d_in[2];
    const float* H   = (const float*)d_in[3];
    const float* lam = (const float*)d_in[4];
    const float* cw  = (const float*)d_in[5]; // [6,3,64,64]
    const float* cb  = (const float*)d_in[6]; // [6,64]
    float* out = (float*)d_out;

    const int E = in_sizes[2];
    const int N = in_sizes[0] / DFEAT;
    const size_t Nf = (size_t)N * DFEAT;

    // carve workspace (256B aligned slices)
    char* w = (char*)d_ws;
    size_t off = 0;
    auto take = [&](size_t bytes) -> char* {
        char* p = w + off;
        off = (off + bytes + 255) & ~(size_t)255;
        return p;
    };
    float*    dinv    = (float*)take((size_t)N * 4);
    unsigned* rowptr  = (unsigned*)take((size_t)(N + 1) * 4);
    unsigned* rowcnt  = (unsigned*)take((size_t)N * 4);
    unsigned* rownext = (unsigned*)take((size_t)N * 4);
    uint2*    csr     = (uint2*)take((size_t)E * 8);
    float*    TX1 = (float*)take(Nf * 4);
    float*    TX2 = (float*)take(Nf * 4);
    float*    TH1 = (float*)take(Nf * 4);   // reused as THR1
    float*    TH2 = (float*)take(Nf * 4);   // reused as THR2
    float*    Zb  = (float*)take(Nf * 4);
    float*    HRb = (float*)take(Nf * 4);
    (void)ws_size; (void)n_in; (void)out_size;

    const int TPB = 256;
    int gN  = (N + TPB - 1) / TPB;
    int gE  = (E + TPB - 1) / TPB;
    int gS  = (int)(((size_t)N * 32 + TPB - 1) / TPB);   // float2 SpMM threads
    int gT  = (N + 15) / 16;

    // 1) degrees + normalization
    zero_u32_kernel<<<gN, TPB, 0, stream>>>((unsigned*)dinv, N);
    deg_kernel<<<gE, TPB, 0, stream>>>(ei, ew, dinv, E);
    dinv_kernel<<<gN, TPB, 0, stream>>>(dinv, N);

    // 2) CSR-by-dst build (count, scan, fill)
    zero_u32_kernel<<<gN, TPB, 0, stream>>>(rowcnt, N);
    count_kernel<<<gE, TPB, 0, stream>>>(ei, rowcnt, E);
    scan_kernel<<<1, 1024, 0, stream>>>(rowcnt, rowptr, rownext, N, E);
    fill_kernel<<<gE, TPB, 0, stream>>>(ei, ew, dinv, rownext, csr, E);

    // 3) Chebyshev bases of X and H (atomic-free gather SpMM, fused combine)
    cheb_spmm_kernel<<<gS, TPB, 0, stream>>>(X,   nullptr, TX1, rowptr, csr, lam, N, 0);
    cheb_spmm_kernel<<<gS, TPB, 0, stream>>>(TX1, X,       TX2, rowptr, csr, lam, N, 1);
    cheb_spmm_kernel<<<gS, TPB, 0, stream>>>(H,   nullptr, TH1, rowptr, csr, lam, N, 0);
    cheb_spmm_kernel<<<gS, TPB, 0, stream>>>(TH1, H,       TH2, rowptr, csr, lam, N, 1);

    // 4) gates Z, R and HR = H*R  (fused 12-term WMMA GEMM)
    gates_kernel<<<gT, 256, 0, stream>>>(X, TX1, TX2, H, TH1, TH2, cw, cb, Zb, HRb, N);

    // 5) Chebyshev basis of HR (reuse TH1/TH2 buffers)
    cheb_spmm_kernel<<<gS, TPB, 0, stream>>>(HRb, nullptr, TH1, rowptr, csr, lam, N, 0);
    cheb_spmm_kernel<<<gS, TPB, 0, stream>>>(TH1, HRb,     TH2, rowptr, csr, lam, N, 1);

    // 6) candidate state + GRU blend (fused 6-term WMMA GEMM)
    final_kernel<<<gT, 128, 0, stream>>>(X, TX1, TX2, HRb, TH1, TH2, cw, cb, Zb, H, out, N);
}